// VGAERModel_46583215292539
// MI455X (gfx1250) — compile-verified
//
#include <hip/hip_runtime.h>
#include <hip/hip_bf16.h>

typedef __attribute__((ext_vector_type(2))) float v2f;
typedef __attribute__((ext_vector_type(8))) float v8f;

#define TB 256  // 8 waves per block (wave32)

#define WMMA_F32(a, b, c) \
  __builtin_amdgcn_wmma_f32_16x16x4_f32(false, (a), false, (b), (short)0, (c), false, false)

// ---------------------------------------------------------------------------
// fp32 GEMM: C[M,N] = A[M,K] @ B[K,N], row-major, V_WMMA_F32_16X16X4_F32.
// One wave computes a 32x32 tile of C (2x2 grid of 16x16 WMMA tiles):
// 4 independent accumulation chains, A/B fragments reused twice each.
// Fragment layouts per CDNA5 ISA 7.12.2:
//   A 16x4:  lane l -> M = l%16 ; vgpr v holds K = kBase + v + 2*(l/16)
//   B 4x16:  lane l -> N = l%16 ; vgpr v holds K = kBase + v + 2*(l/16)
//   C 16x16: lane l -> N = l%16 ; vgpr v holds M = v + 8*(l/16)
// A-side fragment is a contiguous 8B pair -> single global_load_b64.
// ---------------------------------------------------------------------------
__global__ void gemm_wmma_f32(const float* __restrict__ A,
                              const float* __restrict__ B,
                              float* __restrict__ C,
                              int M, int N, int K) {
  const int gid    = blockIdx.x * blockDim.x + threadIdx.x;
  const int wave   = gid >> 5;
  const int lane   = threadIdx.x & 31;
  const int tilesN = N >> 5;
  const int tm = wave / tilesN;
  const int tn = wave - tm * tilesN;
  if (tm >= (M >> 5)) return;  // wave-uniform guard: EXEC stays all-ones

  const int idx  = lane & 15;
  const int half = lane >> 4;

  const float* Ar0 = A + (size_t)(tm * 32 + idx) * K + 2 * half;  // 8B aligned
  const float* Ar1 = Ar0 + (size_t)16 * K;
  const float* Bc0 = B + (size_t)(2 * half) * N + (tn * 32 + idx);
  const float* Bc1 = Bc0 + 16;

  v8f c00 = {}, c01 = {}, c10 = {}, c11 = {};
#pragma unroll 4
  for (int k = 0; k < K; k += 4) {
    v2f a0 = *(const v2f*)(Ar0 + k);       // global_load_b64
    v2f a1 = *(const v2f*)(Ar1 + k);       // global_load_b64
    v2f b0, b1;                            // stride-N rows: two b32 each
    b0.x = Bc0[(size_t)(k + 0) * N]; b0.y = Bc0[(size_t)(k + 1) * N];
    b1.x = Bc1[(size_t)(k + 0) * N]; b1.y = Bc1[(size_t)(k + 1) * N];
    c00 = WMMA_F32(a0, b0, c00);
    c01 = WMMA_F32(a0, b1, c01);
    c10 = WMMA_F32(a1, b0, c10);
    c11 = WMMA_F32(a1, b1, c11);
  }

  float* Cp = C + (size_t)(tm * 32 + 8 * half) * N + tn * 32 + idx;
#pragma unroll
  for (int v = 0; v < 8; ++v) {
    Cp[(size_t)v * N]             = c00[v];
    Cp[(size_t)v * N + 16]        = c01[v];
    Cp[(size_t)(v + 16) * N]      = c10[v];
    Cp[(size_t)(v + 16) * N + 16] = c11[v];
  }
}

// ---------------------------------------------------------------------------
// adj = sigmoid(Z @ Z^T), symmetric. One wave per 32x32 tile of the UPPER
// triangle (tm <= tn): computes the tile (4 WMMA chains, K=64), applies
// sigmoid, stores it coalesced at (tm,tn), then mirrors it to (tn,tm) via a
// padded-LDS transpose so the mirrored store is also 128B-coalesced.
// Both A and B fragments are contiguous Z-row pairs -> global_load_b64 only:
// 4 loads + 4 WMMAs per k-step. WMMA accumulation order is identical for
// (m,n) and (n,m) -> tiles are bit-exact transposes, so diagonal tiles may
// store twice harmlessly. Z is 2 MB -> fully L2-resident (192 MB L2).
// ---------------------------------------------------------------------------
__global__ void zzt_sigmoid_wmma(const float* __restrict__ Z,
                                 float* __restrict__ Out,
                                 int Nn, int H) {
  __shared__ float lds[TB / 32][32][33];  // pad 33: bank-conflict-free transpose

  const int gid   = blockIdx.x * blockDim.x + threadIdx.x;
  const int wave  = gid >> 5;
  const int wInB  = threadIdx.x >> 5;
  const int lane  = threadIdx.x & 31;
  const int tiles = Nn >> 5;              // 256
  const int tm = wave / tiles;
  const int tn = wave - tm * tiles;
  if (tm > tn) return;                    // symmetry: upper triangle only (wave-uniform)

  const int idx  = lane & 15;
  const int half = lane >> 4;

  const float* Ar0 = Z + (size_t)(tm * 32 + idx) * H + 2 * half;  // 8B aligned
  const float* Ar1 = Ar0 + (size_t)16 * H;
  const float* Br0 = Z + (size_t)(tn * 32 + idx) * H + 2 * half;  // B[k][n] = Z[n][k]
  const float* Br1 = Br0 + (size_t)16 * H;

  v8f c00 = {}, c01 = {}, c10 = {}, c11 = {};
#pragma unroll
  for (int k = 0; k < 64; k += 4) {
    v2f a0 = *(const v2f*)(Ar0 + k);      // global_load_b64 x4
    v2f a1 = *(const v2f*)(Ar1 + k);
    v2f b0 = *(const v2f*)(Br0 + k);
    v2f b1 = *(const v2f*)(Br1 + k);
    c00 = WMMA_F32(a0, b0, c00);
    c01 = WMMA_F32(a0, b1, c01);
    c10 = WMMA_F32(a1, b0, c10);
    c11 = WMMA_F32(a1, b1, c11);
  }

  // fused sigmoid
#pragma unroll
  for (int v = 0; v < 8; ++v) {
    c00[v] = 1.0f / (1.0f + __expf(-c00[v]));
    c01[v] = 1.0f / (1.0f + __expf(-c01[v]));
    c10[v] = 1.0f / (1.0f + __expf(-c10[v]));
    c11[v] = 1.0f / (1.0f + __expf(-c11[v]));
  }

  // direct (coalesced) store of tile (tm,tn)
  float* Op = Out + (size_t)(tm * 32 + 8 * half) * Nn + tn * 32 + idx;
#pragma unroll
  for (int v = 0; v < 8; ++v) {
    Op[(size_t)v * Nn]             = c00[v];
    Op[(size_t)v * Nn + 16]        = c01[v];
    Op[(size_t)(v + 16) * Nn]      = c10[v];
    Op[(size_t)(v + 16) * Nn + 16] = c11[v];
  }

  // mirrored store of tile (tn,tm): transpose through LDS, then coalesced rows.
  // lds[nn][mm] = tile[mm][nn]; per-wave region, LDS is in-order per wave.
  {
    const int mm = 8 * half;      // row base within tile for this lane's vgprs
    const int nn = idx;           // col within tile
#pragma unroll
    for (int v = 0; v < 8; ++v) {
      lds[wInB][nn][mm + v]           = c00[v];
      lds[wInB][nn + 16][mm + v]      = c01[v];
      lds[wInB][nn][mm + v + 16]      = c10[v];
      lds[wInB][nn + 16][mm + v + 16] = c11[v];
    }
  }
  asm volatile("" ::: "memory");  // keep ds_store before ds_load (same wave, HW in-order)

  float* Tp = Out + (size_t)(tn * 32) * Nn + tm * 32 + lane;
#pragma unroll
  for (int r = 0; r < 32; ++r)
    Tp[(size_t)r * Nn] = lds[wInB][r][lane];
}

// ---------------------------------------------------------------------------
// COO SpMM scatter: Acc[row[e], :] += val[e] * S[col[e], :]
// One thread per (edge, 4-float chunk); F is a power of two.
// ---------------------------------------------------------------------------
__global__ void spmm_atomic(const int* __restrict__ row,
                            const int* __restrict__ col,
                            const float* __restrict__ val,
                            const float* __restrict__ S,
                            float* __restrict__ Acc,
                            int E, int F) {
  const int t = blockIdx.x * blockDim.x + threadIdx.x;
  const int chunks = F >> 2;
  const int e  = t / chunks;
  if (e >= E) return;
  const int ch = t - e * chunks;

  const int   r  = row[e];
  const int   cI = col[e];
  const float v  = val[e];

  const float4 s = ((const float4*)(S + (size_t)cI * F))[ch];
  float* a = Acc + (size_t)r * F + ch * 4;
  atomicAdd(a + 0, v * s.x);
  atomicAdd(a + 1, v * s.y);
  atomicAdd(a + 2, v * s.z);
  atomicAdd(a + 3, v * s.w);
}

__global__ void zero_f32(float* __restrict__ p, int n) {
  int t = blockIdx.x * blockDim.x + threadIdx.x;
  if (t < n) p[t] = 0.0f;
}

// X[i] = tanh(X[i] + b[i & mask])   (mask = F-1, F power of two)
__global__ void bias_tanh(float* __restrict__ X, const float* __restrict__ b,
                          int total, int mask) {
  int t = blockIdx.x * blockDim.x + threadIdx.x;
  if (t < total) X[t] = tanhf(X[t] + b[t & mask]);
}

// z = mean + noise * exp(log_std)
__global__ void reparam(const float* __restrict__ mean,
                        const float* __restrict__ logstd,
                        const float* __restrict__ noise,
                        float* __restrict__ z, int n) {
  int t = blockIdx.x * blockDim.x + threadIdx.x;
  if (t < n) z[t] = mean[t] + noise[t] * __expf(logstd[t]);
}

extern "C" void kernel_launch(void* const* d_in, const int* in_sizes, int n_in,
                              void* d_out, int out_size, void* d_ws, size_t ws_size,
                              hipStream_t stream) {
  const int N = 8192, E = 262144, IN = 512, H1 = 256, H2 = 64;

  const int*   edge_row = (const int*)d_in[0];
  const int*   edge_col = (const int*)d_in[1];
  const float* edge_val = (const float*)d_in[2];
  const float* features = (const float*)d_in[3];
  const float* W1 = (const float*)d_in[4];
  const float* b1 = (const float*)d_in[5];
  const float* W2 = (const float*)d_in[6];
  const float* b2 = (const float*)d_in[7];
  const float* W3 = (const float*)d_in[8];
  const float* b3 = (const float*)d_in[9];
  const float* noise = (const float*)d_in[10];

  float* out = (float*)d_out;             // adj_rec [N*N] then z [N*H2]
  float* z   = out + (size_t)N * N;

  float* ws   = (float*)d_ws;             // ~24 MB total
  float* sup1 = ws;                       // [N, H1]
  float* agg1 = sup1 + (size_t)N * H1;    // [N, H1] -> becomes h (in-place tanh)
  float* sup2 = agg1 + (size_t)N * H1;    // [N, H2]
  float* sup3 = sup2 + (size_t)N * H2;    // [N, H2]
  float* agg2 = sup3 + (size_t)N * H2;    // [N, H2] -> mean
  float* agg3 = agg2 + (size_t)N * H2;    // [N, H2] -> log_std

  // ---- layer 1: support1 = X @ W1 ; agg1 = A_hat @ support1 ; h = tanh(+b1)
  {
    int waves = (N / 32) * (H1 / 32);               // 2048 waves, 32x32 tiles
    gemm_wmma_f32<<<waves * 32 / TB, TB, 0, stream>>>(features, W1, sup1, N, H1, IN);
  }
  zero_f32<<<(N * H1) / TB, TB, 0, stream>>>(agg1, N * H1);
  spmm_atomic<<<(E * (H1 / 4)) / TB, TB, 0, stream>>>(edge_row, edge_col, edge_val,
                                                      sup1, agg1, E, H1);
  bias_tanh<<<(N * H1) / TB, TB, 0, stream>>>(agg1, b1, N * H1, H1 - 1);

  // ---- layer 2: mean & log_std branches (K = H1 = 256, N = H2 = 64)
  {
    int waves = (N / 32) * (H2 / 32);               // 512 waves
    gemm_wmma_f32<<<waves * 32 / TB, TB, 0, stream>>>(agg1, W2, sup2, N, H2, H1);
    gemm_wmma_f32<<<waves * 32 / TB, TB, 0, stream>>>(agg1, W3, sup3, N, H2, H1);
  }
  zero_f32<<<(2 * N * H2) / TB, TB, 0, stream>>>(agg2, 2 * N * H2);  // agg2+agg3 contiguous
  spmm_atomic<<<(E * (H2 / 4)) / TB, TB, 0, stream>>>(edge_row, edge_col, edge_val,
                                                      sup2, agg2, E, H2);
  spmm_atomic<<<(E * (H2 / 4)) / TB, TB, 0, stream>>>(edge_row, edge_col, edge_val,
                                                      sup3, agg3, E, H2);
  bias_tanh<<<(N * H2) / TB, TB, 0, stream>>>(agg2, b2, N * H2, H2 - 1);
  bias_tanh<<<(N * H2) / TB, TB, 0, stream>>>(agg3, b3, N * H2, H2 - 1);

  // ---- z = mean + noise * exp(log_std)  (written straight into d_out tail)
  reparam<<<(N * H2) / TB, TB, 0, stream>>>(agg2, agg3, noise, z, N * H2);

  // ---- adj_rec = sigmoid(z @ z^T): 256x256 grid of 32x32 tiles; waves with
  //      tm > tn exit immediately (upper-triangle compute + LDS-mirrored store)
  {
    int waves = (N / 32) * (N / 32);                // 65536 waves
    zzt_sigmoid_wmma<<<waves * 32 / TB, TB, 0, stream>>>(z, out, N, H2);
  }
}